// IWHT2Layer_15522011808173
// MI455X (gfx1250) — compile-verified
//
#include <hip/hip_runtime.h>

typedef __attribute__((ext_vector_type(2))) float v2f;
typedef __attribute__((ext_vector_type(8))) float v8f;

// Problem constants (from reference setup_inputs)
constexpr int Tt  = 16;                 // WHT coefficients (4x4)
constexpr int Bsz = 16;                 // batch
constexpr int nH  = 28, nW = 28;        // block grid
constexpr int Cc  = 64;                 // in channels
constexpr int Kk  = 64;                 // out channels per stream (K1 == K2 == 64)
constexpr int Mrows = Bsz * nH * nW;    // 12544 flattened pixels
constexpr int OH = nH * 4;              // 112
constexpr int OW = nW * 4;              // 112
constexpr long OUT_PER_STREAM = (long)Bsz * OH * OW * Kk;

// Fused: per 16-row x 16-col tile, 16 coefficient GEMMs via v_wmma_f32_16x16x4_f32
// with double-buffered register fragments (loads for t+1 in flight during the
// WMMA chain for t), then register-resident 4x4 inverse WHT butterflies,
// scale 1/16 + bias, scatter-store to (B,112,112,K).
__global__ __launch_bounds__(128) void iwht2_fused_kernel(
    const float* __restrict__ trans1, const float* __restrict__ trans2,
    const float* __restrict__ wgt1,   const float* __restrict__ wgt2,
    const float* __restrict__ bias1,  const float* __restrict__ bias2,
    float* __restrict__ out)
{
    const int s = blockIdx.y;                       // stream 0/1
    const float* __restrict__ trans = (s == 0) ? trans1 : trans2;
    const float* __restrict__ wgt   = (s == 0) ? wgt1   : wgt2;
    const float* __restrict__ bias  = (s == 0) ? bias1  : bias2;
    float* __restrict__ o = out + (size_t)s * OUT_PER_STREAM;

    const int lane = threadIdx.x & 31;              // wave32
    const int wave = threadIdx.x >> 5;              // 0..3 -> K tile
    const int n0   = wave * 16;                     // k-column base
    const int m0   = blockIdx.x * 16;               // pixel-row base
    const int r    = lane & 15;
    const int half = lane >> 4;

    // A frag: lane(half,r) holds trans[row=m0+r][c = 4*ks + 2*half + {0,1}]  -> v2f
    const float* aBase = trans + (size_t)(m0 + r) * Cc + 2 * half;
    // B frag: lane(half,r) holds wgt[c = 4*ks + 2*half + {0,1}][k = n0 + r]
    const float* bBase = wgt + (size_t)(2 * half) * Kk + n0 + r;

    v8f acc[16];
    v2f afrag[2][16];                               // double-buffered A fragments
    v2f bfrag[2][16];                               // double-buffered B fragments

    auto loadFrags = [&](int t, int buf) {
        const float* ap = aBase + (size_t)t * Mrows * Cc;
        const float* bp = bBase + (size_t)t * Cc * Kk;
#pragma unroll
        for (int ks = 0; ks < 16; ++ks) {
            afrag[buf][ks] = *(const v2f*)(ap + 4 * ks);      // 8B aligned b64
            v2f b;
            b[0] = bp[(size_t)(4 * ks) * Kk];
            b[1] = bp[(size_t)(4 * ks) * Kk + Kk];
            bfrag[buf][ks] = b;
        }
    };

    loadFrags(0, 0);                                // prologue: fill buffer 0

#pragma unroll
    for (int t = 0; t < 16; ++t) {
        // Issue next coefficient's 48 loads into the other buffer *before*
        // consuming the current one -> WMMAs overlap with memory.
        if (t + 1 < 16) loadFrags(t + 1, (t + 1) & 1);
        if (t + 2 < 16) {
            // pull t+2's A-tile toward the caches (global_prefetch_b8)
            __builtin_prefetch(aBase + (size_t)(t + 2) * Mrows * Cc, 0, 0);
        }
        const int cur = t & 1;
        v8f a = {0.f, 0.f, 0.f, 0.f, 0.f, 0.f, 0.f, 0.f};
#pragma unroll
        for (int ks = 0; ks < 16; ++ks) {
            a = __builtin_amdgcn_wmma_f32_16x16x4_f32(
                    /*neg_a=*/false, afrag[cur][ks], /*neg_b=*/false, bfrag[cur][ks],
                    /*c_mod=*/(short)0, a, /*reuse_a=*/false, /*reuse_b=*/false);
        }
        acc[t] = a;                                 // mixed[t] tile (row g+8*half, col r)
    }

    // Inverse 2D WHT over coefficient index t = 4*u + v, elementwise on v8f.
    // Stage 1: transform over u (H4 rows are +-1 -> radix-4 butterflies).
    v8f st1[16];
#pragma unroll
    for (int v = 0; v < 4; ++v) {
        v8f s0 = acc[0 + v] + acc[8 + v];
        v8f s1 = acc[4 + v] + acc[12 + v];
        v8f d0 = acc[0 + v] - acc[8 + v];
        v8f d1 = acc[4 + v] - acc[12 + v];
        st1[0 * 4 + v] = s0 + s1;   // i = 0
        st1[1 * 4 + v] = s0 - s1;   // i = 1
        st1[2 * 4 + v] = d0 + d1;   // i = 2
        st1[3 * 4 + v] = d0 - d1;   // i = 3
    }
    // Stage 2: transform over v for each spatial row i.
#pragma unroll
    for (int i = 0; i < 4; ++i) {
        v8f x0 = st1[i * 4 + 0], x1 = st1[i * 4 + 1];
        v8f x2 = st1[i * 4 + 2], x3 = st1[i * 4 + 3];
        v8f s0 = x0 + x2, s1 = x1 + x3;
        v8f d0 = x0 - x2, d1 = x1 - x3;
        acc[i * 4 + 0] = s0 + s1;   // j = 0
        acc[i * 4 + 1] = s0 - s1;   // j = 1
        acc[i * 4 + 2] = d0 + d1;   // j = 2
        acc[i * 4 + 3] = d0 - d1;   // j = 3
    }

    const float bv = bias[n0 + r];

    // Scatter: D-matrix reg g holds pixel-row m0 + g + 8*half, lane r holds k = n0 + r.
#pragma unroll
    for (int g = 0; g < 8; ++g) {
        const int m   = m0 + 8 * half + g;
        const int b   = m / (nH * nW);
        const int rem = m - b * (nH * nW);
        const int h   = rem / nW;
        const int w   = rem - h * nW;
        float* obase = o + (((size_t)b * OH + 4 * h) * OW + 4 * w) * Kk + n0 + r;
#pragma unroll
        for (int i = 0; i < 4; ++i) {
#pragma unroll
            for (int j = 0; j < 4; ++j) {
                obase[((size_t)i * OW + j) * Kk] = acc[i * 4 + j][g] * 0.0625f + bv;
            }
        }
    }
}

extern "C" void kernel_launch(void* const* d_in, const int* in_sizes, int n_in,
                              void* d_out, int out_size, void* d_ws, size_t ws_size,
                              hipStream_t stream) {
    const float* trans1 = (const float*)d_in[0];
    const float* trans2 = (const float*)d_in[1];
    const float* wgt1   = (const float*)d_in[2];
    const float* wgt2   = (const float*)d_in[3];
    const float* bias1  = (const float*)d_in[4];
    const float* bias2  = (const float*)d_in[5];
    float* out = (float*)d_out;

    dim3 grid(Mrows / 16, 2);   // 784 pixel-row tiles x 2 streams
    dim3 block(128);            // 4 waves; wave w owns k-columns [16w, 16w+16)
    iwht2_fused_kernel<<<grid, block, 0, stream>>>(trans1, trans2, wgt1, wgt2,
                                                   bias1, bias2, out);
}